// LightGCN_14654428413980
// MI455X (gfx1250) — compile-verified
//
#include <hip/hip_runtime.h>

#define DD 64

typedef __attribute__((ext_vector_type(2))) float v2f;
typedef __attribute__((ext_vector_type(8))) float v8f;

// ---------------------------------------------------------------------------
// x = concat(user_emb, (item_emb + feat0[idx0] + feat1[idx1]) * 1/3)
// out_acc = ALPHA * x
// ---------------------------------------------------------------------------
__global__ void k_init(const float* __restrict__ user_emb,
                       const float* __restrict__ item_emb,
                       const int*   __restrict__ fi0,
                       const int*   __restrict__ fi1,
                       const float* __restrict__ fe0,
                       const float* __restrict__ fe1,
                       float* __restrict__ x,
                       float* __restrict__ out_acc,
                       int NU, int N, float alpha, float wf) {
  int idx = blockIdx.x * blockDim.x + threadIdx.x;
  if (idx >= N * DD) return;
  int i = idx / DD;
  int d = idx - i * DD;
  float v;
  if (i < NU) {
    v = user_emb[idx];
  } else {
    int j = i - NU;
    v = wf * (item_emb[(size_t)j * DD + d] +
              fe0[(size_t)fi0[j] * DD + d] +
              fe1[(size_t)fi1[j] * DD + d]);
  }
  x[idx]       = v;
  out_acc[idx] = alpha * v;
}

// ---------------------------------------------------------------------------
// deg[dst[e]] += 1  (integer-valued float sum -> order-independent, exact)
// ---------------------------------------------------------------------------
__global__ void k_deg(const int* __restrict__ dst, float* __restrict__ deg, int E) {
  int e = blockIdx.x * blockDim.x + threadIdx.x;
  if (e >= E) return;
  __hip_atomic_fetch_add(&deg[dst[e]], 1.0f, __ATOMIC_RELAXED,
                         __HIP_MEMORY_SCOPE_AGENT);
}

__global__ void k_dinv(const float* __restrict__ deg, float* __restrict__ dinv, int N) {
  int i = blockIdx.x * blockDim.x + threadIdx.x;
  if (i >= N) return;
  float d = deg[i];
  dinv[i] = d > 0.0f ? rsqrtf(d) : 0.0f;
}

// ---------------------------------------------------------------------------
// One wave32 per edge: xout[dst] += norm * xin[src]   (float2 per lane)
// L2-resident gather + native global_atomic_add_f32 scatter.
// ---------------------------------------------------------------------------
__global__ void k_scatter(const int* __restrict__ src, const int* __restrict__ dst,
                          const float* __restrict__ dinv,
                          const float* __restrict__ xin,
                          float* __restrict__ xout, int E) {
  int wave = blockIdx.x * (blockDim.x >> 5) + (threadIdx.x >> 5);
  int lane = threadIdx.x & 31;
  if (wave >= E) return;
  if (lane == 0 && wave + 512 < E) {
    __builtin_prefetch(&src[wave + 512], 0, 0);   // global_prefetch_b8
    __builtin_prefetch(&dst[wave + 512], 0, 0);
  }
  int s = src[wave];
  int t = dst[wave];
  float norm = dinv[s] * dinv[t];
  v2f v = *(const v2f*)(xin + (size_t)s * DD + lane * 2);
  float* o = xout + (size_t)t * DD + lane * 2;
  __hip_atomic_fetch_add(o,     v.x * norm, __ATOMIC_RELAXED, __HIP_MEMORY_SCOPE_AGENT);
  __hip_atomic_fetch_add(o + 1, v.y * norm, __ATOMIC_RELAXED, __HIP_MEMORY_SCOPE_AGENT);
}

__global__ void k_accum(float* __restrict__ out_acc, const float* __restrict__ xnew,
                        float alpha, int total) {
  int idx = blockIdx.x * blockDim.x + threadIdx.x;
  if (idx >= total) return;
  out_acc[idx] += alpha * xnew[idx];
}

// ---------------------------------------------------------------------------
// Edge scoring via V_WMMA_F32_16X16X4_F32: one wave computes a 16x16 Gram
// tile D = A x B (A[m][k] = out[src_m][k], B[k][n] = out[dst_n][k]) over
// K=64 as 16 chained k=4 WMMAs, then extracts the diagonal.
//
// 32-bit A 16x4 layout (ISA 7.12.2): lanes 0-15 hold M=lane, K={0,1} in
// vgpr0/1; lanes 16-31 hold K={2,3}. B (4x16) mirrors this with lane = N.
// So every lane loads one float2 per matrix per k-chunk:
//   col = k0 + (lane>>4)*2, row = src/dst[base + (lane&15)].
// C/D layout: VGPR j: lanes 0-15 -> (M=j, N=lane); lanes 16-31 -> (M=8+j,
// N=lane-16). Diagonal: m<8 at lane m vgpr m; m>=8 at lane m+16 vgpr m-8.
// ---------------------------------------------------------------------------
__global__ void k_score(const int* __restrict__ src, const int* __restrict__ dst,
                        const float* __restrict__ out_acc,
                        float* __restrict__ score, int E) {
  int wave = blockIdx.x * (blockDim.x >> 5) + (threadIdx.x >> 5);
  int lane = threadIdx.x & 31;
  long base = (long)wave * 16;
  if (base >= E) return;            // wave-uniform: EXEC stays all-ones below

  if (base + 16 <= E) {             // full 16-edge group (wave-uniform branch)
    int lo = lane & 15;
    int hi = lane >> 4;             // 0 or 1: selects K pair {0,1} vs {2,3}
    int s = src[base + lo];
    int t = dst[base + lo];
    const float* arow = out_acc + (size_t)s * DD;
    const float* brow = out_acc + (size_t)t * DD;

    v8f c = {};
#pragma unroll
    for (int kc = 0; kc < 16; ++kc) {
      int k0 = kc * 4 + hi * 2;
      v2f a = *(const v2f*)(arow + k0);
      v2f b = *(const v2f*)(brow + k0);
      // (neg_a, A, neg_b, B, c_mod, C, reuse_a, reuse_b)
      c = __builtin_amdgcn_wmma_f32_16x16x4_f32(
            false, a, false, b, (short)0, c, false, false);
    }

    float dval = 0.0f;
#pragma unroll
    for (int j = 0; j < 8; ++j)
      if (lane == j || lane == 24 + j) dval = c[j];

    if (lane < 8)        score[base + lane]      = dval;
    else if (lane >= 24) score[base + lane - 16] = dval;
  } else {
    // wave-uniform tail path: scalar 64-length dot per lane
    long e = base + lane;
    if (e < E) {
      int s = src[e];
      int t = dst[e];
      const float* a = out_acc + (size_t)s * DD;
      const float* b = out_acc + (size_t)t * DD;
      float acc = 0.0f;
#pragma unroll
      for (int k = 0; k < DD; ++k) acc = fmaf(a[k], b[k], acc);
      score[e] = acc;
    }
  }
}

// ---------------------------------------------------------------------------
extern "C" void kernel_launch(void* const* d_in, const int* in_sizes, int n_in,
                              void* d_out, int out_size, void* d_ws, size_t ws_size,
                              hipStream_t stream) {
  const int*   edge = (const int*)d_in[0];
  const int*   fi0  = (const int*)d_in[1];
  const int*   fi1  = (const int*)d_in[2];
  const float* ue   = (const float*)d_in[3];
  const float* ie   = (const float*)d_in[4];
  const float* fe0  = (const float*)d_in[5];
  const float* fe1  = (const float*)d_in[6];
  float* score = (float*)d_out;

  const int E  = in_sizes[0] / 2;         // edge_index is [2, E] row-major
  const int NU = in_sizes[3] / DD;
  const int NI = in_sizes[4] / DD;
  const int N  = NU + NI;
  const int NUM_LAYERS = 3;
  const float alpha = 1.0f / (float)(NUM_LAYERS + 1);
  const float wf    = 1.0f / 3.0f;

  const int* srcs = edge;
  const int* dsts = edge + E;

  // Workspace layout: x0 | x1 | out_acc | deg | dinv   (~116.4 MB, L2-resident)
  float* ws = (float*)d_ws;
  size_t nd = (size_t)N * DD;
  float* x0   = ws;
  float* x1   = ws + nd;
  float* oac  = ws + 2 * nd;
  float* deg  = ws + 3 * nd;
  float* dinv = deg + N;

  int total = (int)nd;
  k_init<<<(total + 255) / 256, 256, 0, stream>>>(ue, ie, fi0, fi1, fe0, fe1,
                                                  x0, oac, NU, N, alpha, wf);

  hipMemsetAsync(deg, 0, (size_t)N * sizeof(float), stream);
  k_deg<<<(E + 255) / 256, 256, 0, stream>>>(dsts, deg, E);
  k_dinv<<<(N + 255) / 256, 256, 0, stream>>>(deg, dinv, N);

  float* xin  = x0;
  float* xout = x1;
  for (int l = 0; l < NUM_LAYERS; ++l) {
    hipMemsetAsync(xout, 0, nd * sizeof(float), stream);
    // 8 waves per 256-thread block, one wave per edge
    unsigned blocks = (unsigned)((E + 7) / 8);
    k_scatter<<<blocks, 256, 0, stream>>>(srcs, dsts, dinv, xin, xout, E);
    k_accum<<<(total + 255) / 256, 256, 0, stream>>>(oac, xout, alpha, total);
    float* tmp = xin; xin = xout; xout = tmp;
  }

  long groups = ((long)E + 15) / 16;      // 16 edges per wave
  unsigned sblocks = (unsigned)((groups + 7) / 8);
  k_score<<<sblocks, 256, 0, stream>>>(srcs, dsts, oac, score, E);
}